// SSIMLoss_70695161692380
// MI455X (gfx1250) — compile-verified
//
#include <hip/hip_runtime.h>

// ---------------------------------------------------------------------------
// SSIM loss, fused single pass for MI455X (gfx1250, wave32).
// Memory-bound: 100.7 MB in -> 1 scalar out; roofline ~4.3us @ 23.3 TB/s.
// - async global->LDS staging of x/y tiles (+5px halo), zero-padded edges
// - separable 11x11 box filter: horizontal sliding sums (5 planes in LDS),
//   vertical sliding sums in registers, fused SSIM math
// - block reduction finishes on the matrix pipe (v_wmma_f32_16x16x4_f32,
//   A = ones => column sums), then one f64 global atomic per block
// ---------------------------------------------------------------------------

#define WIN   11
#define RAD   5
#define OUT_H 32
#define OUT_W 64
#define IN_H  (OUT_H + WIN - 1)   // 42
#define IN_W  (OUT_W + WIN - 1)   // 74
#define XS_STRIDE 75              // odd stride -> conflict-free row access
#define HS_STRIDE 65              // odd stride -> conflict-free plane access
#define IMG_H 512
#define IMG_W 512
#define NPLANES 48                // B*C = 16*3
#define NPIX (16LL * 3 * 512 * 512)

typedef float v2f __attribute__((ext_vector_type(2)));
typedef float v8f __attribute__((ext_vector_type(8)));

__global__ void ssim_zero(double* acc) { *acc = 0.0; }

__global__ __launch_bounds__(256) void ssim_fused(const float* __restrict__ X,
                                                  const float* __restrict__ Y,
                                                  double* __restrict__ acc) {
  __shared__ float xs[IN_H * XS_STRIDE];          // 12600 B
  __shared__ float ys[IN_H * XS_STRIDE];          // 12600 B
  __shared__ float hs[5 * IN_H * HS_STRIDE];      // 54600 B
  float* red = xs;  // xs is dead after phase B; reuse as reduction buffer

  const int tid = threadIdx.x;
  const int x0 = blockIdx.x * OUT_W - RAD;
  const int y0 = blockIdx.y * OUT_H - RAD;
  const size_t plane = (size_t)blockIdx.z * (IMG_H * IMG_W);
  const float* __restrict__ Xp = X + plane;
  const float* __restrict__ Yp = Y + plane;

  // ---- Phase A: async global->LDS stage of the halo tile (zeros for OOB) --
  const unsigned xs_base = (unsigned)(size_t)(void*)xs;
  const unsigned ys_base = (unsigned)(size_t)(void*)ys;
  for (int i = tid; i < IN_H * IN_W; i += 256) {
    int r = i / IN_W;
    int c = i - r * IN_W;
    int gy = y0 + r;
    int gx = x0 + c;
    int lidx = r * XS_STRIDE + c;
    if ((unsigned)gy < (unsigned)IMG_H && (unsigned)gx < (unsigned)IMG_W) {
      unsigned voff = (unsigned)(gy * IMG_W + gx) * 4u;
      unsigned lx = xs_base + (unsigned)lidx * 4u;
      unsigned ly = ys_base + (unsigned)lidx * 4u;
      asm volatile("global_load_async_to_lds_b32 %0, %1, %2"
                   :: "v"(lx), "v"(voff), "s"(Xp) : "memory");
      asm volatile("global_load_async_to_lds_b32 %0, %1, %2"
                   :: "v"(ly), "v"(voff), "s"(Yp) : "memory");
    } else {
      xs[lidx] = 0.0f;
      ys[lidx] = 0.0f;
    }
  }
  asm volatile("s_wait_asynccnt 0" ::: "memory");
  __syncthreads();

  // ---- Phase B: horizontal 11-tap sliding sums, 5 quantities ----------
  // task = (quantity q in 0..4, row r in 0..41); 210 active threads
  if (tid < 5 * IN_H) {
    const int q = tid / IN_H;
    const int r = tid - q * IN_H;
    const float* __restrict__ xr = xs + r * XS_STRIDE;
    const float* __restrict__ yr = ys + r * XS_STRIDE;
    float* __restrict__ hq = hs + (q * IN_H + r) * HS_STRIDE;

    auto val = [&](int i) -> float {
      float a = xr[i], b = yr[i];
      return (q == 0) ? a
           : (q == 1) ? b
           : (q == 2) ? a * a
           : (q == 3) ? b * b
           :            a * b;
    };

    float s = 0.0f;
#pragma unroll
    for (int i = 0; i < WIN - 1; ++i) s += val(i);
    for (int c = 0; c < OUT_W; ++c) {
      s += val(c + WIN - 1);
      hq[c] = s;
      s -= val(c);
    }
  }
  __syncthreads();

  // ---- Phase C: vertical 11-tap sliding sums in registers + SSIM -------
  // thread = (column c in 0..63, row-strip rg in 0..3 of 8 rows)
  const int c  = tid & 63;
  const int r0 = (tid >> 6) * 8;

  float sums[5];
  float lacc = 0.0f;
#pragma unroll
  for (int q = 0; q < 5; ++q) {
    const float* __restrict__ p = hs + (q * IN_H + r0) * HS_STRIDE + c;
    float s = 0.0f;
#pragma unroll
    for (int w = 0; w < WIN - 1; ++w) s += p[w * HS_STRIDE];
    sums[q] = s;
  }

  const float inv = 1.0f / (float)(WIN * WIN);
  const float C1 = 0.01f * 0.01f;
  const float C2 = 0.03f * 0.03f;
  const float EPS = 1e-8f;

#pragma unroll
  for (int rr = 0; rr < 8; ++rr) {
    const int r = r0 + rr;
#pragma unroll
    for (int q = 0; q < 5; ++q)
      sums[q] += hs[(q * IN_H + r + WIN - 1) * HS_STRIDE + c];

    float mux = sums[0] * inv;
    float muy = sums[1] * inv;
    float mux2 = mux * mux;
    float muy2 = muy * muy;
    float muxy = mux * muy;
    float sx  = sums[2] * inv - mux2;
    float sy  = sums[3] * inv - muy2;
    float sxy = sums[4] * inv - muxy;
    float num = (2.0f * muxy + C1) * (2.0f * sxy + C2);
    float den = (mux2 + muy2 + C1) * (sx + sy + C2) + EPS;
    lacc += num / den;

#pragma unroll
    for (int q = 0; q < 5; ++q)
      sums[q] -= hs[(q * IN_H + r) * HS_STRIDE + c];
  }

  // ---- Phase D: block reduction; final 32->1 on the matrix pipe --------
  red[tid] = lacc;
  __syncthreads();

  if (tid < 32) {           // wave 0, EXEC all ones (WMMA requirement)
    float p = red[tid];
#pragma unroll
    for (int i = 1; i < 8; ++i) p += red[tid + i * 32];

    // D = A(ones,16x4) * B(partials,4x16): every lane's d[0] is a column
    // sum over K; the 32 lane values of B vgpr0 cover all partials, vgpr1=0.
    v2f a; a[0] = 1.0f; a[1] = 1.0f;
    v2f b; b[0] = p;    b[1] = 0.0f;
    v8f cz = {};
    v8f d = __builtin_amdgcn_wmma_f32_16x16x4_f32(
        false, a, false, b, (short)0, cz, false, false);

    float t = d[0];
    t += __shfl_xor(t, 8);
    t += __shfl_xor(t, 4);
    t += __shfl_xor(t, 2);
    t += __shfl_xor(t, 1);
    if (tid == 0) atomicAdd(acc, (double)t);
  }
}

__global__ void ssim_fin(float* out, const double* acc) {
  out[0] = (float)(1.0 - acc[0] / (double)NPIX);
}

extern "C" void kernel_launch(void* const* d_in, const int* in_sizes, int n_in,
                              void* d_out, int out_size, void* d_ws, size_t ws_size,
                              hipStream_t stream) {
  (void)in_sizes; (void)n_in; (void)out_size; (void)ws_size;
  const float* x = (const float*)d_in[0];
  const float* y = (const float*)d_in[1];
  float* out = (float*)d_out;
  double* acc = (double*)d_ws;

  ssim_zero<<<1, 1, 0, stream>>>(acc);

  dim3 grid(IMG_W / OUT_W, IMG_H / OUT_H, NPLANES);  // 8 x 16 x 48
  dim3 block(256);
  ssim_fused<<<grid, block, 0, stream>>>(x, y, acc);

  ssim_fin<<<1, 1, 0, stream>>>(out, acc);
}